// LinearEncoder_54924041781477
// MI455X (gfx1250) — compile-verified
//
#include <hip/hip_runtime.h>

typedef __attribute__((ext_vector_type(2))) float v2f;
typedef __attribute__((ext_vector_type(8))) float v8f;

#define IN_C  128
#define OUT_C 64

// ---------------------------------------------------------------------------
// Kernel 1: init degree (self-loop contributes 1 to every node)
// ---------------------------------------------------------------------------
__global__ void gcn_deg_init(float* __restrict__ deg, int n) {
    int i = blockIdx.x * blockDim.x + threadIdx.x;
    if (i < n) deg[i] = 1.0f;
}

// ---------------------------------------------------------------------------
// Kernel 2: degree accumulation over edge destinations
// ---------------------------------------------------------------------------
__global__ void gcn_deg_accum(const int* __restrict__ dst, float* __restrict__ deg, int e) {
    int i = blockIdx.x * blockDim.x + threadIdx.x;
    if (i < e) unsafeAtomicAdd(&deg[dst[i]], 1.0f);
}

// ---------------------------------------------------------------------------
// Kernel 3: dis = rsqrt(deg)   (deg >= 1 always, so no zero guard needed)
// ---------------------------------------------------------------------------
__global__ void gcn_rsqrt(float* __restrict__ dis, int n) {
    int i = blockIdx.x * blockDim.x + threadIdx.x;
    if (i < n) dis[i] = rsqrtf(dis[i]);
}

// ---------------------------------------------------------------------------
// Kernel 4: h = x @ W^T  via V_WMMA_F32_16X16X4_F32 (full fp32 precision).
// One wave computes a 16-row x 64-col stripe of h.
//   A (16x4 f32): lanes 0-15 = rows, VGPR0 = K{0|2}, VGPR1 = K{1|3} by lane half
//   B (4x16 f32): lanes 0-15 = cols, same K striping  (B[k][n] = W[n][k])
//   C/D (16x16):  VGPR r -> row r (lanes 0-15) / row r+8 (lanes 16-31)
// ---------------------------------------------------------------------------
__global__ __launch_bounds__(256)
void gcn_gemm_wmma(const float* __restrict__ x, const float* __restrict__ W,
                   float* __restrict__ h, int n) {
    const int lane = threadIdx.x & 31;
    const int wave = threadIdx.x >> 5;
    const int half = lane >> 4;     // 0: K={0,1}, 1: K={2,3}
    const int l16  = lane & 15;

    const int m0 = (blockIdx.x * 8 + wave) * 16;
    if (m0 >= n) return;            // wave-uniform: EXEC stays all-ones

    int row = m0 + l16;
    if (row >= n) row = n - 1;      // clamp keeps EXEC full; tail stores guarded

    const float* xp  = x + (size_t)row * IN_C + 2 * half;
    const float* wp0 = W + (size_t)(l16 +  0) * IN_C + 2 * half;
    const float* wp1 = W + (size_t)(l16 + 16) * IN_C + 2 * half;
    const float* wp2 = W + (size_t)(l16 + 32) * IN_C + 2 * half;
    const float* wp3 = W + (size_t)(l16 + 48) * IN_C + 2 * half;

    v8f acc0 = {}, acc1 = {}, acc2 = {}, acc3 = {};

#pragma unroll
    for (int k = 0; k < IN_C; k += 4) {
        v2f a  = *(const v2f*)(xp  + k);
        v2f b0 = *(const v2f*)(wp0 + k);
        v2f b1 = *(const v2f*)(wp1 + k);
        v2f b2 = *(const v2f*)(wp2 + k);
        v2f b3 = *(const v2f*)(wp3 + k);
        acc0 = __builtin_amdgcn_wmma_f32_16x16x4_f32(false, a, false, b0, (short)0, acc0, false, false);
        acc1 = __builtin_amdgcn_wmma_f32_16x16x4_f32(false, a, false, b1, (short)0, acc1, false, false);
        acc2 = __builtin_amdgcn_wmma_f32_16x16x4_f32(false, a, false, b2, (short)0, acc2, false, false);
        acc3 = __builtin_amdgcn_wmma_f32_16x16x4_f32(false, a, false, b3, (short)0, acc3, false, false);
    }

    float* hp = h + (size_t)m0 * OUT_C;
#pragma unroll
    for (int r = 0; r < 8; ++r) {
        int m = r + 8 * half;
        if (m0 + m < n) {
            float* rp = hp + (size_t)m * OUT_C + l16;
            rp[0]  = acc0[r];
            rp[16] = acc1[r];
            rp[32] = acc2[r];
            rp[48] = acc3[r];
        }
    }
}

// ---------------------------------------------------------------------------
// Kernel 5: out = h * dis^2 + b   (self-loop term; fully overwrites d_out)
// One thread handles 4 channels (float4).
// ---------------------------------------------------------------------------
__global__ void gcn_self_bias(const float* __restrict__ h, const float* __restrict__ dis,
                              const float* __restrict__ b, float* __restrict__ out, int n) {
    int idx = blockIdx.x * blockDim.x + threadIdx.x;   // over n * 16
    if (idx >= n * 16) return;
    int i = idx >> 4;
    int q = idx & 15;
    float d  = dis[i];
    float s  = d * d;
    float4 hv = ((const float4*)h)[idx];
    float4 bv = ((const float4*)b)[q];
    float4 o;
    o.x = hv.x * s + bv.x;
    o.y = hv.y * s + bv.y;
    o.z = hv.z * s + bv.z;
    o.w = hv.w * s + bv.w;
    ((float4*)out)[idx] = o;
}

// ---------------------------------------------------------------------------
// Kernel 6: edge scatter: out[dst] += h[src] * dis[src]*dis[dst]
// 16 threads per edge, float4 gather + 4 hardware f32 atomics (L2-resident).
// ---------------------------------------------------------------------------
__global__ void gcn_scatter(const int* __restrict__ src, const int* __restrict__ dst,
                            const float* __restrict__ h, const float* __restrict__ dis,
                            float* __restrict__ out, int e) {
    int idx = blockIdx.x * blockDim.x + threadIdx.x;   // over e * 16
    if (idx >= e * 16) return;
    int ed = idx >> 4;
    int q  = idx & 15;
    int s  = src[ed];
    int d  = dst[ed];
    float norm = dis[s] * dis[d];
    float4 hv = ((const float4*)h)[(size_t)s * 16 + q];
    float* op = out + (size_t)d * OUT_C + q * 4;
    unsafeAtomicAdd(op + 0, hv.x * norm);
    unsafeAtomicAdd(op + 1, hv.y * norm);
    unsafeAtomicAdd(op + 2, hv.z * norm);
    unsafeAtomicAdd(op + 3, hv.w * norm);
}

// ---------------------------------------------------------------------------
extern "C" void kernel_launch(void* const* d_in, const int* in_sizes, int n_in,
                              void* d_out, int out_size, void* d_ws, size_t ws_size,
                              hipStream_t stream) {
    const float* x  = (const float*)d_in[0];
    const int*   ei = (const int*)d_in[1];       // (2, E) row-major int32
    const float* W  = (const float*)d_in[2];     // (64, 128)
    const float* b  = (const float*)d_in[3];     // (64,)

    const int n = in_sizes[0] / IN_C;            // 100000
    const int e = in_sizes[1] / 2;               // 1600000
    const int* src = ei;
    const int* dst = ei + e;

    float* out = (float*)d_out;
    float* h   = (float*)d_ws;                   // n * 64 floats
    float* dis = h + (size_t)n * OUT_C;          // n floats

    const int T = 256;

    // 1) degree init (self loop)
    gcn_deg_init<<<(n + T - 1) / T, T, 0, stream>>>(dis, n);
    // 2) degree accumulation
    gcn_deg_accum<<<(e + T - 1) / T, T, 0, stream>>>(dst, dis, e);
    // 3) dis = rsqrt(deg)
    gcn_rsqrt<<<(n + T - 1) / T, T, 0, stream>>>(dis, n);
    // 4) h = x @ W^T  (WMMA fp32)
    {
        int tiles  = (n + 15) / 16;
        int blocks = (tiles + 7) / 8;            // 8 waves per 256-thread block
        gcn_gemm_wmma<<<blocks, T, 0, stream>>>(x, W, h, n);
    }
    // 5) out = h * dis^2 + b
    gcn_self_bias<<<(n * 16 + T - 1) / T, T, 0, stream>>>(h, dis, b, out, n);
    // 6) scatter edges
    gcn_scatter<<<((size_t)e * 16 + T - 1) / T, T, 0, stream>>>(src, dst, h, dis, out, e);
}